// DualGraphFusionGCN_45543833207100
// MI455X (gfx1250) — compile-verified
//
#include <hip/hip_runtime.h>
#include <hip/hip_bf16.h>

typedef __attribute__((ext_vector_type(2))) float v2f;
typedef __attribute__((ext_vector_type(8))) float v8f;

#define IN_DIM 7
#define H1 64
#define H2 128
#define N_CLS 9
#define N_DOM 2
#define MT 4            // M-tiles (of 16 rows) per wave in the WMMA GEMM

// ---------------------------------------------------------------- utilities

__global__ __launch_bounds__(256) void k_fill(float* __restrict__ p, float v, int n) {
    int i = blockIdx.x * 256 + threadIdx.x;
    if (i < n) p[i] = v;
}

__global__ __launch_bounds__(256) void k_edge_deg(const int* __restrict__ col,
                                                  float* __restrict__ deg, int E) {
    int e = blockIdx.x * 256 + threadIdx.x;
    if (e < E) atomicAdd(&deg[col[e]], 1.0f);
}

__global__ __launch_bounds__(256) void k_rsqrt(float* __restrict__ p, int n) {
    int i = blockIdx.x * 256 + threadIdx.x;
    if (i < n) p[i] = rsqrtf(p[i]);   // deg >= 1 always (self loop)
}

// h = x @ W for the K=7 input transform (too small for WMMA)
__global__ __launch_bounds__(256) void k_xform7(const float* __restrict__ x,
                                                const float* __restrict__ Wm,
                                                float* __restrict__ h,
                                                int total, int lgF) {
    int idx = blockIdx.x * 256 + threadIdx.x;
    if (idx >= total) return;
    int F = 1 << lgF;
    int n = idx >> lgF, f = idx & (F - 1);
    const float* xr = x + (size_t)n * IN_DIM;
    float acc = 0.f;
#pragma unroll
    for (int k = 0; k < IN_DIM; ++k) acc = fmaf(xr[k], Wm[k * F + f], acc);
    h[idx] = acc;
}

// out[i] = h[i] * dis[i]^2   (self-loop contribution, also zero-inits out)
__global__ __launch_bounds__(256) void k_selfloop(const float* __restrict__ h,
                                                  const float* __restrict__ dis,
                                                  float* __restrict__ out,
                                                  int total, int lgF) {
    int idx = blockIdx.x * 256 + threadIdx.x;
    if (idx >= total) return;
    int n = idx >> lgF;
    float d = dis[n];
    out[idx] = h[idx] * d * d;
}

// out[col[e]] += h[row[e]] * dis[row]*dis[col], 4 features per thread
__global__ __launch_bounds__(256) void k_edge_agg(const int* __restrict__ row,
                                                  const int* __restrict__ colv,
                                                  const float* __restrict__ dis,
                                                  const float* __restrict__ h,
                                                  float* __restrict__ out,
                                                  long long total, int lgC4) {
    long long idx = (long long)blockIdx.x * 256 + threadIdx.x;
    if (idx >= total) return;
    int c4 = 1 << lgC4;
    long long e = idx >> lgC4;
    int c = (int)(idx & (c4 - 1));
    int r = row[e], t = colv[e];
    float nrm = dis[r] * dis[t];
    const float4 v = *(const float4*)(h + (((size_t)r) << (lgC4 + 2)) + 4 * c);
    float* o = out + (((size_t)t) << (lgC4 + 2)) + 4 * c;
    atomicAdd(o + 0, v.x * nrm);
    atomicAdd(o + 1, v.y * nrm);
    atomicAdd(o + 2, v.z * nrm);
    atomicAdd(o + 3, v.w * nrm);
}

__global__ __launch_bounds__(256) void k_bias_relu(float* __restrict__ h,
                                                   const float* __restrict__ b,
                                                   long long total, int lgF) {
    long long idx = (long long)blockIdx.x * 256 + threadIdx.x;
    if (idx >= total) return;
    int f = (int)(idx & ((1 << lgF) - 1));
    float v = h[idx] + b[f];
    h[idx] = v > 0.f ? v : 0.f;
}

// ---------------------------------------------------------------- fp32 WMMA GEMM
// C[M x N] (+)= A[M x K] @ B[K x N], optional bias add + relu.
// Wave computes a 64x16 slab: MT=4 independent 16x16 accumulators sharing each
// B fragment (V_WMMA_F32_16X16X4_F32, exact fp32). 4 A b64 loads + 2 B b32
// loads per 4 WMMAs; 4 independent D->C chains hide XDL latency.
// blockDim.x = 32*(N/16); grid.x = ceil((M/16)/MT). Requires M%16==0, K%4==0.
__global__ __launch_bounds__(256) void k_wmma_gemm(const float* __restrict__ A,
                                                   const float* __restrict__ B,
                                                   float* __restrict__ C,
                                                   const float* __restrict__ bias,
                                                   int M, int N, int K,
                                                   int lda, int ldb, int ldc,
                                                   int accumulate, int do_relu) {
    const int lane = threadIdx.x & 31;
    const int wave = threadIdx.x >> 5;
    const int half = lane >> 4;   // K-pair selector: lanes 0-15 -> K 0..1, 16-31 -> K 2..3
    const int lr   = lane & 15;   // row index (A) / column index (B, C/D)
    const int n0 = wave * 16;
    if (n0 >= N) return;

    const int ntm   = M >> 4;           // number of 16-row tiles
    const int tile0 = blockIdx.x * MT;
    if (tile0 >= ntm) return;

    // Clamp out-of-range tiles to the last valid tile (redundant compute);
    // stores are guarded by wave-uniform predicates so EXEC stays all-ones.
    const float* Arow[MT];
    size_t crow[MT];
    int valid[MT];
#pragma unroll
    for (int i = 0; i < MT; ++i) {
        int t = tile0 + i;
        valid[i] = (t < ntm);
        int tc = valid[i] ? t : (ntm - 1);
        Arow[i] = A + (size_t)(tc * 16 + lr) * lda + 2 * half;
        crow[i] = (size_t)(tc * 16);
        __builtin_prefetch(Arow[i], 0, 1);   // global_prefetch_b8
    }

    v8f acc[MT];
#pragma unroll
    for (int i = 0; i < MT; ++i) {
        if (accumulate) {
#pragma unroll
            for (int v = 0; v < 8; ++v)
                acc[i][v] = C[(crow[i] + v + 8 * half) * ldc + n0 + lr];
        } else {
#pragma unroll
            for (int v = 0; v < 8; ++v) acc[i][v] = 0.0f;
        }
    }

    const float* Bcol = B + (size_t)(2 * half) * ldb + n0 + lr;

    for (int k = 0; k < K; k += 4) {
        v2f b;
        b.x = Bcol[(size_t)k * ldb];
        b.y = Bcol[(size_t)(k + 1) * ldb];
#pragma unroll
        for (int i = 0; i < MT; ++i) {
            v2f a;
            a.x = Arow[i][k];
            a.y = Arow[i][k + 1];
            acc[i] = __builtin_amdgcn_wmma_f32_16x16x4_f32(
                /*neg_a=*/false, a, /*neg_b=*/false, b,
                /*c_mod=*/(short)0, acc[i], /*reuse_a=*/false, /*reuse_b=*/false);
        }
    }

    const float bv = bias ? bias[n0 + lr] : 0.0f;
#pragma unroll
    for (int i = 0; i < MT; ++i) {
        if (!valid[i]) continue;   // uniform across the wave
#pragma unroll
        for (int v = 0; v < 8; ++v) {
            float r = acc[i][v] + bv;
            if (do_relu) r = r > 0.f ? r : 0.f;
            C[(crow[i] + v + 8 * half) * ldc + n0 + lr] = r;
        }
    }
}

// ---------------------------------------------------------------- fusion head
// One wave per node: logits = hidden @ A2 + ab2 (K=128, 2 outputs),
// softmax over 2, fused = w0*xs + w1*xc_dec.
__global__ __launch_bounds__(256) void k_softmax_fuse(const float* __restrict__ hidden,
                                                      const float* __restrict__ A2,
                                                      const float* __restrict__ ab2,
                                                      const float* __restrict__ xs,
                                                      const float* __restrict__ xcd,
                                                      float* __restrict__ fused, int N) {
    int warp = (blockIdx.x * 256 + threadIdx.x) >> 5;
    int lane = threadIdx.x & 31;
    if (warp >= N) return;
    const float* hr = hidden + (size_t)warp * H2;
    float p0 = 0.f, p1 = 0.f;
#pragma unroll
    for (int f = lane; f < H2; f += 32) {
        float hv = hr[f];
        p0 = fmaf(hv, A2[f * 2 + 0], p0);
        p1 = fmaf(hv, A2[f * 2 + 1], p1);
    }
#pragma unroll
    for (int off = 16; off > 0; off >>= 1) {
        p0 += __shfl_xor(p0, off, 32);
        p1 += __shfl_xor(p1, off, 32);
    }
    float l0 = p0 + ab2[0], l1 = p1 + ab2[1];
    float m = fmaxf(l0, l1);
    float e0 = expf(l0 - m), e1 = expf(l1 - m);
    float inv = 1.f / (e0 + e1);
    float w0 = e0 * inv, w1 = e1 * inv;
    const float* xr = xs + (size_t)warp * H2;
    const float* cr = xcd + (size_t)warp * H2;
    float* fr = fused + (size_t)warp * H2;
#pragma unroll
    for (int f = lane; f < H2; f += 32) fr[f] = w0 * xr[f] + w1 * cr[f];
}

// out[n,f] = b[f] + sum_k X[n,k] * Wm[k,f]   (small F heads: 9 / 2 columns)
__global__ __launch_bounds__(256) void k_head_small(const float* __restrict__ X,
                                                    const float* __restrict__ Wm,
                                                    const float* __restrict__ b,
                                                    float* __restrict__ out,
                                                    int N, int K, int F) {
    int idx = blockIdx.x * 256 + threadIdx.x;
    if (idx >= N * F) return;
    int n = idx / F, f = idx % F;
    const float* xr = X + (size_t)n * K;
    float acc = b[f];
    for (int k = 0; k < K; ++k) acc = fmaf(xr[k], Wm[k * F + f], acc);
    out[idx] = acc;
}

// ---------------------------------------------------------------- launch

static inline unsigned gdiv(long long n) { return (unsigned)((n + 255) / 256); }
static inline unsigned ggemm(int M) { return (unsigned)(((M >> 4) + MT - 1) / MT); }

extern "C" void kernel_launch(void* const* d_in, const int* in_sizes, int n_in,
                              void* d_out, int out_size, void* d_ws, size_t ws_size,
                              hipStream_t stream) {
    const float* x_s  = (const float*)d_in[0];
    const float* x_c  = (const float*)d_in[1];
    const float* Q    = (const float*)d_in[2];
    const int*   ei_s = (const int*)d_in[3];
    const int*   ei_c = (const int*)d_in[4];
    const float* W1s  = (const float*)d_in[5];
    const float* b1s  = (const float*)d_in[6];
    const float* W2s  = (const float*)d_in[7];
    const float* b2s  = (const float*)d_in[8];
    const float* W1c  = (const float*)d_in[9];
    const float* b1c  = (const float*)d_in[10];
    const float* W2c  = (const float*)d_in[11];
    const float* b2c  = (const float*)d_in[12];
    const float* A1   = (const float*)d_in[13];
    const float* ab1  = (const float*)d_in[14];
    const float* A2   = (const float*)d_in[15];
    const float* ab2  = (const float*)d_in[16];
    const float* Wcls = (const float*)d_in[17];
    const float* bcls = (const float*)d_in[18];
    const float* D1   = (const float*)d_in[19];
    const float* db1  = (const float*)d_in[20];
    const float* D2   = (const float*)d_in[21];
    const float* db2  = (const float*)d_in[22];

    const int Ns = in_sizes[0] / IN_DIM;     // 100000
    const int Nc = in_sizes[1] / IN_DIM;     // 512
    const int Es = in_sizes[3] / 2;          // 3200000
    const int Ec = in_sizes[4] / 2;          // 16384
    const int* row_s = ei_s, * col_s = ei_s + Es;
    const int* row_c = ei_c, * col_c = ei_c + Ec;

    // workspace layout (floats), with aggressive aliasing; peak ~154 MB
    float* W = (float*)d_ws;
    size_t o = 0;
    float* dis_s = W + o; o += (size_t)((Ns + 31) & ~31);
    float* dis_c = W + o; o += 512;
    float* hc1   = W + o; o += (size_t)Nc * H1;
    float* ac1   = W + o; o += (size_t)Nc * H1;
    float* hc2   = W + o; o += (size_t)Nc * H2;
    float* xcf   = W + o; o += (size_t)Nc * H2;
    float* R1    = W + o; o += (size_t)Ns * H1;   // h1_s -> hidden(lo) -> dtmp
    float* R2    = W + o; o += (size_t)Ns * H1;   // xs1  -> hidden(hi)
    float* R3    = W + o; o += (size_t)Ns * H2;   // h2_s -> xc_dec
    float* R4    = W + o; o += (size_t)Ns * H2;   // xs
    float* hidden = R1;                            // Ns x 128 spanning R1+R2
    float* dtmp   = R1;                            // Ns x 64 after hidden dies

    float* out_cls = (float*)d_out;
    float* out_dom = out_cls + (size_t)Ns * N_CLS;
    float* fused   = out_dom + (size_t)Ns * N_DOM;

    // ---- symmetric-normalization degrees (self loops folded as init=1)
    k_fill<<<gdiv(Ns), 256, 0, stream>>>(dis_s, 1.0f, Ns);
    k_edge_deg<<<gdiv(Es), 256, 0, stream>>>(col_s, dis_s, Es);
    k_rsqrt<<<gdiv(Ns), 256, 0, stream>>>(dis_s, Ns);
    k_fill<<<gdiv(Nc), 256, 0, stream>>>(dis_c, 1.0f, Nc);
    k_edge_deg<<<gdiv(Ec), 256, 0, stream>>>(col_c, dis_c, Ec);
    k_rsqrt<<<gdiv(Nc), 256, 0, stream>>>(dis_c, Nc);

    // ---- sample branch, layer 1 (7 -> 64)
    k_xform7<<<gdiv((long long)Ns * H1), 256, 0, stream>>>(x_s, W1s, R1, Ns * H1, 6);
    k_selfloop<<<gdiv((long long)Ns * H1), 256, 0, stream>>>(R1, dis_s, R2, Ns * H1, 6);
    k_edge_agg<<<gdiv((long long)Es * (H1 / 4)), 256, 0, stream>>>(row_s, col_s, dis_s, R1, R2,
                                                                   (long long)Es * (H1 / 4), 4);
    k_bias_relu<<<gdiv((long long)Ns * H1), 256, 0, stream>>>(R2, b1s, (long long)Ns * H1, 6);

    // ---- sample branch, layer 2 (64 -> 128), transform via fp32 WMMA
    k_wmma_gemm<<<dim3(ggemm(Ns)), dim3(256), 0, stream>>>(R2, W2s, R3, nullptr,
                                                           Ns, 128, 64, 64, 128, 128, 0, 0);
    k_selfloop<<<gdiv((long long)Ns * H2), 256, 0, stream>>>(R3, dis_s, R4, Ns * H2, 7);
    k_edge_agg<<<gdiv((long long)Es * (H2 / 4)), 256, 0, stream>>>(row_s, col_s, dis_s, R3, R4,
                                                                   (long long)Es * (H2 / 4), 5);
    k_bias_relu<<<gdiv((long long)Ns * H2), 256, 0, stream>>>(R4, b2s, (long long)Ns * H2, 7);

    // ---- cluster branch (tiny)
    k_xform7<<<gdiv((long long)Nc * H1), 256, 0, stream>>>(x_c, W1c, hc1, Nc * H1, 6);
    k_selfloop<<<gdiv((long long)Nc * H1), 256, 0, stream>>>(hc1, dis_c, ac1, Nc * H1, 6);
    k_edge_agg<<<gdiv((long long)Ec * (H1 / 4)), 256, 0, stream>>>(row_c, col_c, dis_c, hc1, ac1,
                                                                   (long long)Ec * (H1 / 4), 4);
    k_bias_relu<<<gdiv((long long)Nc * H1), 256, 0, stream>>>(ac1, b1c, (long long)Nc * H1, 6);
    k_wmma_gemm<<<dim3(ggemm(Nc)), dim3(256), 0, stream>>>(ac1, W2c, hc2, nullptr,
                                                           Nc, 128, 64, 64, 128, 128, 0, 0);
    k_selfloop<<<gdiv((long long)Nc * H2), 256, 0, stream>>>(hc2, dis_c, xcf, Nc * H2, 7);
    k_edge_agg<<<gdiv((long long)Ec * (H2 / 4)), 256, 0, stream>>>(row_c, col_c, dis_c, hc2, xcf,
                                                                   (long long)Ec * (H2 / 4), 5);
    k_bias_relu<<<gdiv((long long)Nc * H2), 256, 0, stream>>>(xcf, b2c, (long long)Nc * H2, 7);

    // ---- xc_dec = Q @ xc   (M=100000, K=512, N=128)  -- dominant GEMM, fp32 WMMA
    k_wmma_gemm<<<dim3(ggemm(Ns)), dim3(256), 0, stream>>>(Q, xcf, R3, nullptr,
                                                           Ns, 128, 512, 512, 128, 128, 0, 0);

    // ---- hidden = relu([xs | xc_dec] @ A1 + ab1), split-K over the concat
    k_wmma_gemm<<<dim3(ggemm(Ns)), dim3(256), 0, stream>>>(R4, A1, hidden, nullptr,
                                                           Ns, 128, 128, 128, 128, 128, 0, 0);
    k_wmma_gemm<<<dim3(ggemm(Ns)), dim3(256), 0, stream>>>(R3, A1 + 128 * 128, hidden, ab1,
                                                           Ns, 128, 128, 128, 128, 128, 1, 1);

    // ---- gate softmax + fused blend (writes straight into d_out's fused slot)
    k_softmax_fuse<<<gdiv((long long)Ns * 32), 256, 0, stream>>>(hidden, A2, ab2, R4, R3,
                                                                 fused, Ns);

    // ---- class head (N=9, scalar) ; domain head via fp32 WMMA then scalar tail
    k_head_small<<<gdiv((long long)Ns * N_CLS), 256, 0, stream>>>(fused, Wcls, bcls, out_cls,
                                                                  Ns, H2, N_CLS);
    k_wmma_gemm<<<dim3(ggemm(Ns)), dim3(128), 0, stream>>>(fused, D1, dtmp, db1,
                                                           Ns, 64, 128, 128, 64, 64, 0, 1);
    k_head_small<<<gdiv((long long)Ns * N_DOM), 256, 0, stream>>>(dtmp, D2, db2, out_dom,
                                                                  Ns, H1, N_DOM);
}